// Encoder_11330123727469
// MI455X (gfx1250) — compile-verified
//
#include <hip/hip_runtime.h>
#include <hip/hip_bf16.h>

typedef __attribute__((ext_vector_type(2))) float v2f;
typedef __attribute__((ext_vector_type(8))) float v8f;

#define BB   64      // batch
#define TT   512     // seq len
#define EE   256     // embed dim
#define UU   512     // hidden
#define GG   2048    // 4*U gates
#define MM   (BB*TT) // 32768 rows
#define SCAN_BLOCKS 16

__device__ __forceinline__ v8f wmma4(v2f a, v2f b, v8f c) {
  // D = A(16x4 f32) * B(4x16 f32) + C(16x16 f32), full f32 WMMA
  return __builtin_amdgcn_wmma_f32_16x16x4_f32(false, a, false, b, (short)0, c,
                                               false, false);
}

__device__ __forceinline__ float sigmoidf_(float v) {
  return 1.0f / (1.0f + __expf(-v));
}

// ---------------------------------------------------------------------------
// init: zero h ping-pong buffer and grid-barrier state (ws is poisoned 0xAA)
// ---------------------------------------------------------------------------
__global__ void lstm_init_kernel(float* __restrict__ hbuf, int n,
                                 int* __restrict__ bar) {
  int i = blockIdx.x * blockDim.x + threadIdx.x;
  if (i < n) hbuf[i] = 0.0f;
  if (i < 2) bar[i] = 0;
}

// ---------------------------------------------------------------------------
// x-projection GEMM: P[r, 0:2048] = A_row(r) @ W + bias
//   gather=1: A_row(r) = W_emb[idx[r]]  (K = E)
//   gather=0: A_row(r) = A[r]           (K = U)
// One wave computes a 16x64 tile via 4 f32 WMMA accumulators.
// grid: 8192 blocks x 256 threads -> 65536 waves = 2048 mtiles * 32 nquads
// ---------------------------------------------------------------------------
__global__ void xproj_gemm_kernel(const float* __restrict__ Abase,
                                  const int* __restrict__ idx, int gather,
                                  int K, const float* __restrict__ W,
                                  const float* __restrict__ bias,
                                  float* __restrict__ P) {
  const int tid  = threadIdx.x;
  const int lane = tid & 31;
  const int wid  = blockIdx.x * (blockDim.x >> 5) + (tid >> 5);
  const int nq    = wid & 31;        // 0..31 -> 64-col quad
  const int mtile = wid >> 5;        // 0..2047
  const int m  = lane & 15;
  const int hi = lane >> 4;
  const int rowbase = mtile * 16;
  const int colbase = nq * 64;

  const int arw = rowbase + m;
  const float* arow = gather ? (Abase + (size_t)idx[arw] * K)
                             : (Abase + (size_t)arw * K);

  v8f acc[4];
  acc[0] = v8f{0,0,0,0,0,0,0,0};
  acc[1] = v8f{0,0,0,0,0,0,0,0};
  acc[2] = v8f{0,0,0,0,0,0,0,0};
  acc[3] = v8f{0,0,0,0,0,0,0,0};

  for (int kk = 0; kk < K; kk += 4) {
    const int kr = kk + 2 * hi;
    v2f a = *(const v2f*)(arow + kr);
    const float* w0 = W + (size_t)kr * GG + colbase + m;
    const float* w1 = w0 + GG;
    v2f b0 = { w0[0],  w1[0]  };
    v2f b1 = { w0[16], w1[16] };
    v2f b2 = { w0[32], w1[32] };
    v2f b3 = { w0[48], w1[48] };
    acc[0] = wmma4(a, b0, acc[0]);
    acc[1] = wmma4(a, b1, acc[1]);
    acc[2] = wmma4(a, b2, acc[2]);
    acc[3] = wmma4(a, b3, acc[3]);
  }

#pragma unroll
  for (int q = 0; q < 4; ++q) {
    const int col = colbase + q * 16 + m;
    const float bv = bias[col];
#pragma unroll
    for (int j = 0; j < 8; ++j) {
      const int r = rowbase + 8 * hi + j;
      P[(size_t)r * GG + col] = acc[q][j] + bv;
    }
  }
}

// ---------------------------------------------------------------------------
// grid-wide barrier (16 co-resident blocks, counter + generation)
// ---------------------------------------------------------------------------
__device__ __forceinline__ void grid_barrier(int* cnt, int* gen, int nb) {
  __syncthreads();
  if (threadIdx.x == 0) {
    __threadfence();
    int g = atomicAdd(gen, 0);
    int old = atomicAdd(cnt, 1);
    if (old == nb - 1) {
      atomicExch(cnt, 0);
      __threadfence();
      atomicAdd(gen, 1);
    } else {
      while (atomicAdd(gen, 0) == g) { __builtin_amdgcn_s_sleep(2); }
    }
    __threadfence();
  }
  __syncthreads();
}

// ---------------------------------------------------------------------------
// persistent LSTM scan: 16 blocks x 256 threads = 128 waves.
// wave (btile 0..3, utile 0..31) owns a 16(batch) x 16(hidden) tile.
// z = P[:,t,:] + h_prev @ Wh ; gates ; c kept in VGPRs across all T steps.
// ---------------------------------------------------------------------------
__global__ void lstm_scan_kernel(const float* __restrict__ P,
                                 const float* __restrict__ Wh,
                                 const int* __restrict__ x,
                                 float* __restrict__ hbuf,
                                 float* __restrict__ hs_out,
                                 float* __restrict__ out_h,
                                 float* __restrict__ out_c,
                                 int* __restrict__ bar_cnt,
                                 int* __restrict__ bar_gen) {
  const int tid  = threadIdx.x;
  const int lane = tid & 31;
  const int wid  = blockIdx.x * (blockDim.x >> 5) + (tid >> 5); // 0..127
  const int btile = wid & 3;
  const int utile = wid >> 2;
  const int m  = lane & 15;
  const int hi = lane >> 4;
  const int rowbase = btile * 16;
  const int ucol = utile * 16 + m;

  v8f c = v8f{0,0,0,0,0,0,0,0};
  float hlast[8];
#pragma unroll
  for (int j = 0; j < 8; ++j) hlast[j] = 0.0f;

  for (int t = 0; t < TT; ++t) {
    const float* hp = hbuf + (size_t)(t & 1) * (BB * UU);
    float*       hn = hbuf + (size_t)((t + 1) & 1) * (BB * UU);

    v8f ai = v8f{0,0,0,0,0,0,0,0};
    v8f af = v8f{0,0,0,0,0,0,0,0};
    v8f ag = v8f{0,0,0,0,0,0,0,0};
    v8f ao = v8f{0,0,0,0,0,0,0,0};

    const float* arow = hp + (size_t)(rowbase + m) * UU;
    for (int kk = 0; kk < UU; kk += 4) {
      const int kr = kk + 2 * hi;
      v2f a = *(const v2f*)(arow + kr);
      const float* w0 = Wh + (size_t)kr * GG + ucol;
      const float* w1 = w0 + GG;
      v2f bi = { w0[0 * UU], w1[0 * UU] };
      v2f bf = { w0[1 * UU], w1[1 * UU] };
      v2f bg = { w0[2 * UU], w1[2 * UU] };
      v2f bo = { w0[3 * UU], w1[3 * UU] };
      ai = wmma4(a, bi, ai);
      af = wmma4(a, bf, af);
      ag = wmma4(a, bg, ag);
      ao = wmma4(a, bo, ao);
    }

#pragma unroll
    for (int j = 0; j < 8; ++j) {
      const int bb = rowbase + 8 * hi + j;
      const float* prow = P + ((size_t)bb * TT + t) * GG;
      const float zi = ai[j] + prow[0 * UU + ucol];
      const float zf = af[j] + prow[1 * UU + ucol];
      const float zg = ag[j] + prow[2 * UU + ucol];
      const float zo = ao[j] + prow[3 * UU + ucol];
      const float iv = sigmoidf_(zi);
      const float fv = sigmoidf_(zf);
      const float gv = tanhf(zg);
      const float ov = sigmoidf_(zo);
      const float cn = fv * c[j] + iv * gv;
      const float hv = ov * tanhf(cn);
      const bool  mm = (x[bb * TT + t] != 0);
      const float hpv = hp[(size_t)bb * UU + ucol];
      const float ho = mm ? hv : hpv;
      c[j] = mm ? cn : c[j];
      hn[(size_t)bb * UU + ucol] = ho;
      if (hs_out) hs_out[((size_t)bb * TT + t) * UU + ucol] = ho;
      hlast[j] = ho;
    }

    grid_barrier(bar_cnt, bar_gen, SCAN_BLOCKS);
  }

#pragma unroll
  for (int j = 0; j < 8; ++j) {
    const int bb = rowbase + 8 * hi + j;
    out_h[(size_t)bb * UU + ucol] = hlast[j];
    out_c[(size_t)bb * UU + ucol] = c[j];
  }
}

// ---------------------------------------------------------------------------
extern "C" void kernel_launch(void* const* d_in, const int* in_sizes, int n_in,
                              void* d_out, int out_size, void* d_ws, size_t ws_size,
                              hipStream_t stream) {
  (void)in_sizes; (void)n_in; (void)out_size; (void)ws_size;
  const int*   x     = (const int*)d_in[0];
  const float* W_emb = (const float*)d_in[1];
  const float* Wx0   = (const float*)d_in[2];
  const float* Wh0   = (const float*)d_in[3];
  const float* b0    = (const float*)d_in[4];
  const float* Wx1   = (const float*)d_in[5];
  const float* Wh1   = (const float*)d_in[6];
  const float* b1    = (const float*)d_in[7];
  float* out = (float*)d_out;

  // workspace carve-up
  float* P    = (float*)d_ws;                         // 32768*2048 f32 (256 MB)
  float* hs   = P  + (size_t)MM * GG;                 // 32768*512  f32 (64 MB)
  float* hbuf = hs + (size_t)MM * UU;                 // 2*64*512   f32 (256 KB)
  int*   bar  = (int*)(hbuf + (size_t)2 * BB * UU);   // [cnt, gen]

  const size_t BU = (size_t)BB * UU;

  // ---- layer 0 ----
  lstm_init_kernel<<<256, 256, 0, stream>>>(hbuf, 2 * BB * UU, bar);
  xproj_gemm_kernel<<<8192, 256, 0, stream>>>(W_emb, x, 1, EE, Wx0, b0, P);
  lstm_scan_kernel<<<SCAN_BLOCKS, 256, 0, stream>>>(
      P, Wh0, x, hbuf, hs, out + 0 * BU, out + 1 * BU, bar, bar + 1);

  // ---- layer 1 ----
  lstm_init_kernel<<<256, 256, 0, stream>>>(hbuf, 2 * BB * UU, bar);
  xproj_gemm_kernel<<<8192, 256, 0, stream>>>(hs, nullptr, 0, UU, Wx1, b1, P);
  lstm_scan_kernel<<<SCAN_BLOCKS, 256, 0, stream>>>(
      P, Wh1, x, hbuf, nullptr, out + 2 * BU, out + 3 * BU, bar, bar + 1);
}